// MultiheadAttention_8495445311912
// MI455X (gfx1250) — compile-verified
//
#include <hip/hip_runtime.h>
#include <hip/hip_bf16.h>
#include <math.h>

// Problem constants (from reference)
#define BN   32
#define KK   2048
#define NH   16
#define CC   64
#define EE   1024          // NH*CC
#define E2   2048          // N_ACT*EE
#define RNORM 0.125f       // 1/sqrt(C) = 1/8

typedef __attribute__((ext_vector_type(2))) float v2f;
typedef __attribute__((ext_vector_type(8))) float v8f;

// ---------------------------------------------------------------------------
// Generic small GEMM: out[M x Nout] = A[M x Kt] * W^T (+ bias), W is [Nout x Kt]
// One wave (32 lanes) computes one 16x16 tile of out using V_WMMA_F32_16X16X4_F32.
//
// VGPR layouts (CDNA5 ISA 7.12.2, 32-bit, wave32):
//   A 16x4:  lanes 0-15 -> M=lane, {K=0,K=1}; lanes 16-31 -> M=lane-16, {K=2,K=3}
//   B 4x16:  lanes 0-15 -> N=lane, {K=0,K=1}; lanes 16-31 -> N=lane-16, {K=2,K=3}
//   C/D 16x16: VGPR r: lanes 0-15 -> (M=r,     N=lane),
//                      lanes 16-31 -> (M=r+8, N=lane-16)
// ---------------------------------------------------------------------------
__global__ void mha_proj_wmma_f32(const float* __restrict__ A,
                                  const float* __restrict__ W,
                                  const float* __restrict__ bias,
                                  float* __restrict__ out,
                                  int M, int Kt, int Nout) {
    const int mtiles = M >> 4;
    const int tile   = blockIdx.x;
    const int mt     = tile % mtiles;
    const int nt     = tile / mtiles;

    const int lane = threadIdx.x & 31;
    const int L    = lane & 15;
    const int hi   = lane >> 4;

    const int arow = mt * 16 + L;   // A row for this lane
    const int wrow = nt * 16 + L;   // W row (== output column) for this lane

    const float* ap = A + (size_t)arow * Kt + 2 * hi;
    const float* wp = W + (size_t)wrow * Kt + 2 * hi;

    v8f acc = {};
    for (int kk = 0; kk < Kt; kk += 4) {
        v2f a, b;
        a.x = ap[kk + 0];
        a.y = ap[kk + 1];
        b.x = wp[kk + 0];
        b.y = wp[kk + 1];
        // D = A*B + C   (8-arg form: neg_a, A, neg_b, B, c_mod, C, reuse_a, reuse_b)
        acc = __builtin_amdgcn_wmma_f32_16x16x4_f32(false, a, false, b,
                                                    (short)0, acc, false, false);
    }

    const float bv = bias[wrow];
#pragma unroll
    for (int r = 0; r < 8; ++r) {
        const int orow = mt * 16 + hi * 8 + r;
        out[(size_t)orow * Nout + wrow] = acc[r] + bv;
    }
}

// ---------------------------------------------------------------------------
// Streaming attention core: one workgroup per (n,h); 8 waves, each wave
// processes keys w, w+8, ... with a flash-attention-style online softmax plus
// the un-normalized tanh*gate coda accumulation. Lane c-pair = (2*lane,2*lane+1).
// ---------------------------------------------------------------------------
__global__ __launch_bounds__(256)
void mha_attn_stream(const float* __restrict__ qp,     // [N, 2048]
                     const float* __restrict__ kmat,   // [N, K, H, C]
                     const float* __restrict__ vmat,   // [N, K, H, C]
                     const unsigned char* __restrict__ msk, // [N, K] (bool)
                     float* __restrict__ mix) {        // [N, E]
    const int n = blockIdx.x >> 4;
    const int h = blockIdx.x & 15;

    const int tid  = threadIdx.x;
    const int lane = tid & 31;
    const int w    = tid >> 5;      // wave id 0..7

    // Per-lane query pair for this head
    const float* qph = qp + (size_t)n * E2 + h * (2 * CC);
    const float2 qs = *(const float2*)(qph + 2 * lane);        // smax half
    const float2 qc = *(const float2*)(qph + CC + 2 * lane);   // coda half

    const size_t headOff = ((size_t)n * KK * NH + h) * CC;
    const float* kb = kmat + headOff + 2 * lane;
    const float* vb = vmat + headOff + 2 * lane;
    const unsigned char* mb = msk + (size_t)n * KK;
    const size_t keyStride = (size_t)NH * CC;   // 1024 floats between keys

    float mrun = -1e30f, lrun = 0.0f;
    float vax = 0.0f, vay = 0.0f;   // softmax-weighted V accumulators
    float cax = 0.0f, cay = 0.0f;   // coda (tanh*gate) accumulators

    for (int kk = w; kk < KK; kk += 8) {
        const float* kp = kb + (size_t)kk * keyStride;
        const float* vp = vb + (size_t)kk * keyStride;
        const float2 kv = *(const float2*)kp;
        const float2 vv = *(const float2*)vp;
        __builtin_prefetch(kp + 8 * keyStride, 0, 1);   // global_prefetch_b8
        __builtin_prefetch(vp + 8 * keyStride, 0, 1);

        float ds = qs.x * kv.x + qs.y * kv.y;
        float dc = qc.x * kv.x + qc.y * kv.y;
        float da = fabsf(qc.x - kv.x) + fabsf(qc.y - kv.y);
#pragma unroll
        for (int off = 16; off; off >>= 1) {
            ds += __shfl_xor(ds, off, 32);
            dc += __shfl_xor(dc, off, 32);
            da += __shfl_xor(da, off, 32);
        }

        const bool mv = mb[kk] != 0;
        const float s  = mv ? ds * RNORM : -1e30f;
        const float t  = tanhf(dc * RNORM);
        const float g  = mv ? 2.0f / (1.0f + __expf(da * RNORM)) : 0.0f;
        const float ac = t * g;

        const float nm = fmaxf(mrun, s);
        const float sc = __expf(mrun - nm);
        const float p  = __expf(s - nm);
        lrun = lrun * sc + p;
        vax  = vax * sc + p * vv.x;
        vay  = vay * sc + p * vv.y;
        cax += ac * vv.x;
        cay += ac * vv.y;
        mrun = nm;
    }

    // Cross-wave merge in LDS
    __shared__ float smx[8], sls[8];
    __shared__ float svv[8][CC];
    __shared__ float scd[8][CC];
    svv[w][2 * lane]     = vax;
    svv[w][2 * lane + 1] = vay;
    scd[w][2 * lane]     = cax;
    scd[w][2 * lane + 1] = cay;
    if (lane == 0) { smx[w] = mrun; sls[w] = lrun; }
    __syncthreads();

    if (tid < CC) {
        float gm = -1e30f;
#pragma unroll
        for (int i = 0; i < 8; ++i) gm = fmaxf(gm, smx[i]);
        float gl = 0.0f, vs = 0.0f, cs = 0.0f;
#pragma unroll
        for (int i = 0; i < 8; ++i) {
            const float e = __expf(smx[i] - gm);
            gl += sls[i] * e;
            vs += svv[i][tid] * e;
            cs += scd[i][tid];
        }
        mix[(size_t)n * EE + h * CC + tid] = 0.5f * (vs / gl + cs);
    }
}

extern "C" void kernel_launch(void* const* d_in, const int* in_sizes, int n_in,
                              void* d_out, int out_size, void* d_ws, size_t ws_size,
                              hipStream_t stream) {
    const float*         q     = (const float*)d_in[0];
    const float*         kmat  = (const float*)d_in[1];
    const float*         vmat  = (const float*)d_in[2];
    const unsigned char* msk   = (const unsigned char*)d_in[3];
    const float*         W_in  = (const float*)d_in[4];
    const float*         b_in  = (const float*)d_in[5];
    const float*         W_out = (const float*)d_in[6];
    const float*         b_out = (const float*)d_in[7];
    float*               out   = (float*)d_out;

    float* qp  = (float*)d_ws;            // [32, 2048]  = 256 KB
    float* mix = qp + (size_t)BN * E2;    // [32, 1024]  = 128 KB

    // 1) qp = q @ W_in.T + b_in    (32x1024 @ 1024x2048)
    {
        const int blocks = (BN / 16) * (E2 / 16);   // 2 * 128 = 256 waves
        mha_proj_wmma_f32<<<blocks, 32, 0, stream>>>(q, W_in, b_in, qp, BN, EE, E2);
    }
    // 2) streaming attention -> mix
    mha_attn_stream<<<BN * NH, 256, 0, stream>>>(qp, kmat, vmat, msk, mix);
    // 3) out = mix @ W_out.T + b_out   (32x1024 @ 1024x1024)
    {
        const int blocks = (BN / 16) * (EE / 16);   // 2 * 64 = 128 waves
        mha_proj_wmma_f32<<<blocks, 32, 0, stream>>>(mix, W_out, b_out, out, BN, EE, EE);
    }
    (void)in_sizes; (void)n_in; (void)out_size; (void)ws_size;
}